// DNS_1236950582115
// MI455X (gfx1250) — compile-verified
//
#include <hip/hip_runtime.h>
#include <stdint.h>

// ---- problem constants (from reference: B=256, T=1024, C=64) ----
#define BB   256
#define TT   1024
#define CC   64
#define TM1  1023   // T-1 (output t extent)

// ---- truncated-parallel-Thomas tiling ----
#define SCH  64     // output t-values per block
#define HF   32     // forward-sweep halo (carry decays <= ~0.4/step)
#define HB   32     // backward-sweep halo
#define HD   16     // diagonal-recurrence warmup halo
#define NCH  16     // ceil(1023/64)

#define NX_MAX   131   // x rows staged: SCH + HF + HB + 3
#define NTDR_MAX 148   // tdr span: HD + HF + SCH + HB + 2
#define NDW_MAX  132   // dinv span: HF + SCH + HB + 1
#define NBF_MAX  97    // forward results kept: SCH + HB + 1

typedef unsigned int u32x4 __attribute__((ext_vector_type(4)));
typedef int          i32x8 __attribute__((ext_vector_type(8)));
typedef int          i32x4 __attribute__((ext_vector_type(4)));

__global__ __launch_bounds__(CC) void spline_coeff_kernel(
    const float* __restrict__ times,  // (B, T)
    const float* __restrict__ x,      // (B, T, C)
    float* __restrict__ out)          // (4, B, T-1, C)
{
  __shared__ float xs[NX_MAX * CC];      // staged x tile (TDM target)
  __shared__ float bf_s[NBF_MAX * CC];   // forward-swept rhs per channel
  __shared__ float tdr_s[NTDR_MAX];      // 1/dt
  __shared__ float dinv_s[NDW_MAX];      // 1/d (LU diagonal)

  const int tid = threadIdx.x;           // lane = channel (coalesced rows)
  const int bx  = blockIdx.x;
  const int b   = bx / NCH;
  const int ch  = bx - b * NCH;

  const int t0    = ch * SCH;
  const int t_end = (t0 + SCH < TM1) ? (t0 + SCH) : TM1;        // outputs [t0, t_end)
  const int t_hi  = (t_end + HB < TT - 1) ? (t_end + HB) : (TT - 1); // bwd seed
  const int t_lo  = (t0 - HF > 0) ? (t0 - HF) : 0;              // fwd start
  const int t_tlo = (t_lo - HD - 1 > 0) ? (t_lo - HD - 1) : 0;  // tdr base
  const int t_xlo = (t_lo - 1 > 0) ? (t_lo - 1) : 0;            // x tile base
  const int t_xhi = (t_hi + 1 < TT - 1) ? (t_hi + 1) : (TT - 1);
  const int nx    = t_xhi - t_xlo + 1;                          // <= 131 rows
  const int t_thi = (t_hi < TT - 2) ? t_hi : (TT - 2);          // last valid tdr
  const int ntdr  = t_thi - t_tlo + 1;
  const int ndw   = t_hi - t_lo + 1;

  // ---------------- TDM: async DMA of x tile (nx rows x 64 f32) into LDS --------
  {
    uint64_t ga  = (uint64_t)(uintptr_t)(x + ((size_t)b * TT + t_xlo) * CC);
    uint32_t lds = (uint32_t)(uintptr_t)(&xs[0]);   // flat LDS addr low32 = offset
    u32x4 g0;
    g0[0] = 1u;                                        // count=1, load descriptor
    g0[1] = lds;                                       // lds_addr
    g0[2] = (uint32_t)ga;                              // global_addr[31:0]
    g0[3] = (uint32_t)((ga >> 32) & 0x1FFFFFFu) | (2u << 30); // addr[56:32] | type=2
    i32x8 g1;
    g1[0] = (int)(2u << 16);                 // data_size = 4 bytes
    g1[1] = (int)(64u << 16);                // tensor_dim0 = 64 (low16 in [31:16])
    g1[2] = (int)((uint32_t)nx << 16);       // tensor_dim1 = nx (low16 in [31:16])
    g1[3] = (int)(64u << 16);                // tile_dim0 = 64
    g1[4] = (int)(uint32_t)nx;               // tile_dim1 = nx, tile_dim2 = 0 (2D)
    g1[5] = 64;                              // tensor_dim0_stride = 64 elements
    g1[6] = 0;
    g1[7] = 0;
    i32x4 g2 = {0, 0, 0, 0};                 // unused (2D tile)
    i32x4 g3 = {0, 0, 0, 0};                 // unused
    i32x8 g4 = {0, 0, 0, 0, 0, 0, 0, 0};     // unused extra group (6-arg builtin)
    __builtin_amdgcn_tensor_load_to_lds(g0, g1, g2, g3, g4, 0);
  }

  // ---------------- tdr = 1/dt, computed while the DMA is in flight -------------
  const float* trow = times + (size_t)b * TT;
  for (int i = tid; i < ntdr; i += CC) {
    int t = t_tlo + i;
    tdr_s[i] = 1.0f / (trow[t + 1] - trow[t]);
  }
  __syncthreads();

  auto tdr_at = [&](int k) -> float {
    return (k >= t_tlo && k <= t_thi) ? tdr_s[k - t_tlo] : 0.0f; // zero-padded
  };

  // ------- LU diagonal 1/d via warm-started truncated recurrence (parallel) -----
  // d[t] = sd[t] - tdr[t-1]^2 / d[t-1],  sd[k] = 2*(tdr[k-1]+tdr[k]).
  // Diagonal dominance => influence of the seed decays ~0.14/step; HD=16 => 2e-14.
  for (int j = tid; j < ndw; j += CC) {
    int t  = t_lo + j;
    int ws = (t - HD > 0) ? (t - HD) : 0;
    float d = 2.0f * (tdr_at(ws - 1) + tdr_at(ws));
    for (int k = ws + 1; k <= t; ++k) {
      float e = tdr_at(k - 1);
      d = 2.0f * (e + tdr_at(k)) - (e * e) / d;
    }
    dinv_s[j] = 1.0f / d;
  }

  __builtin_amdgcn_s_wait_tensorcnt(0);   // x tile resident in LDS
  __syncthreads();                        // dinv_s / tdr_s visible to all

  // ---------------- forward sweep: bf[t] = rhs[t] - w[t]*bf[t-1] ----------------
  // rhs[t] = pds[t-1] + pds[t], pds[t] = 3*(x[t+1]-x[t])*tdr[t]^2, w = tdr*dinv.
  const int c = tid;
  float x_cur = xs[(t_lo - t_xlo) * CC + c];
  float pds_prev = 0.0f;
  if (t_lo >= 1) {
    float e = tdr_s[(t_lo - 1) - t_tlo];
    pds_prev = 3.0f * (x_cur - xs[(t_lo - 1 - t_xlo) * CC + c]) * e * e;
  }
  float bprev = 0.0f, dinv_prev = 0.0f;
  for (int t = t_lo; t <= t_hi; ++t) {
    float pds_cur = 0.0f;
    float x_next  = x_cur;
    if (t <= TT - 2) {
      x_next = xs[(t + 1 - t_xlo) * CC + c];
      float e = tdr_s[t - t_tlo];
      pds_cur = 3.0f * (x_next - x_cur) * e * e;
    }
    float r  = pds_prev + pds_cur;
    float bf = (t == t_lo) ? r : (r - tdr_s[(t - 1) - t_tlo] * dinv_prev * bprev);
    if (t >= t0) bf_s[(t - t0) * CC + c] = bf;
    bprev = bf;
    pds_prev = pds_cur;
    x_cur = x_next;
    dinv_prev = dinv_s[t - t_lo];
  }

  // -------- backward substitution + fused coefficient emission ------------------
  const size_t PL = (size_t)BB * TM1 * CC;  // output plane stride
  float kd_next = bf_s[(t_hi - t0) * CC + c] * dinv_s[t_hi - t_lo]; // exact if t_hi==T-1
  for (int t = t_hi - 1; t >= t0; --t) {
    float up   = tdr_s[t - t_tlo];                                  // t <= T-2 here
    float kd_t = (bf_s[(t - t0) * CC + c] - up * kd_next) * dinv_s[t - t_lo];
    if (t < t_end) {
      float xt  = xs[(t     - t_xlo) * CC + c];
      float xt1 = xs[(t + 1 - t_xlo) * CC + c];
      float sp  = 6.0f * (xt1 - xt) * up;                           // six_pd * tdr
      size_t base = ((size_t)b * TM1 + t) * CC + c;
      out[base]          = xt;                                      // a
      out[PL + base]     = kd_t;                                    // b
      out[2 * PL + base] = (sp - 4.0f * kd_t - 2.0f * kd_next) * up;          // two_c
      out[3 * PL + base] = (-sp + 3.0f * (kd_t + kd_next)) * up * up;         // three_d
    }
    kd_next = kd_t;
  }
}

extern "C" void kernel_launch(void* const* d_in, const int* in_sizes, int n_in,
                              void* d_out, int out_size, void* d_ws, size_t ws_size,
                              hipStream_t stream) {
  (void)in_sizes; (void)n_in; (void)out_size; (void)d_ws; (void)ws_size;
  const float* times = (const float*)d_in[0];  // (B, T) f32
  const float* x     = (const float*)d_in[1];  // (B, T, C) f32
  float* out         = (float*)d_out;          // (4, B, T-1, C) f32

  dim3 grid(BB * NCH);   // 4096 independent (batch, t-chunk) tiles
  dim3 block(CC);        // 64 threads = 2 wave32, lane = channel
  hipLaunchKernelGGL(spline_coeff_kernel, grid, block, 0, stream, times, x, out);
}